// GraphDecoder_56573309223972
// MI455X (gfx1250) — compile-verified
//
#include <hip/hip_runtime.h>
#include <stdint.h>

typedef __attribute__((ext_vector_type(2))) float v2f;
typedef __attribute__((ext_vector_type(8))) float v8f;

// Exact parameter element type expected by the async-to-LDS builtin
// (diagnostic: "__attribute__((__vector_size__(4 * sizeof(int)))) int __device__ *")
typedef int vi4 __attribute__((vector_size(4 * sizeof(int))));
typedef __attribute__((address_space(1))) vi4 g_vi4;   // global (AS1)
typedef __attribute__((address_space(3))) vi4 l_vi4;   // LDS (AS3)

#define N_NODES 50000
#define N_EDGES 800000
#define LATENT 128
#define HIDDEN 256
#define N_NODE_TYPES 32
#define N_EDGE_TYPES 8

// Padded LDS row stride (floats): 260*4B = 1040B (8/16B aligned rows),
// 260 % 64 == 4 -> the 32 lanes of an A-fragment column read hit 32 distinct banks.
#define FSTRIDE 260

// Workspace layout (floats): transposed / padded weights.
#define WS_W1T_OFF 0                                   // [HIDDEN][HIDDEN] = W1^T
#define WS_W2T_OFF (HIDDEN * HIDDEN)                   // [16][HIDDEN]     = padded W2^T
#define WS_WNT_OFF (HIDDEN * HIDDEN + 16 * HIDDEN)     // [32][LATENT]     = W_node^T
#define WS_FLOATS  (HIDDEN * HIDDEN + 16 * HIDDEN + N_NODE_TYPES * LATENT)

#if defined(__has_builtin)
#if __has_builtin(__builtin_amdgcn_global_load_async_to_lds_b128)
#define USE_ASYNC_LDS 1
#endif
#endif

static __device__ __forceinline__ v8f wmma_f32(v2f a, v2f b, v8f c) {
    // V_WMMA_F32_16X16X4_F32 : D = A(16x4) * B(4x16) + C, all f32, RNE.
    return __builtin_amdgcn_wmma_f32_16x16x4_f32(
        /*neg_a=*/false, a, /*neg_b=*/false, b,
        /*c_mod=*/(short)0, c, /*reuse_a=*/false, /*reuse_b=*/false);
}

// ---------------------------------------------------------------------------
// Prep: one-time weight transposes so all B fragments are contiguous b64 loads.
//   W1T[n][k] = W1[k][n]          (256x256)
//   W2T[n][k] = n<8 ? W2[k][n]:0  (16x256, zero-padded N)
//   WnT[n][k] = Wn[k][n]          (32x128)
// ---------------------------------------------------------------------------
__global__ __launch_bounds__(256) void prep_weights_kernel(
    const float* __restrict__ W1, const float* __restrict__ W2,
    const float* __restrict__ Wn, float* __restrict__ ws) {
    const int n = blockIdx.x;     // 0..255
    const int k = threadIdx.x;    // 0..255
    ws[WS_W1T_OFF + n * HIDDEN + k] = W1[(size_t)k * HIDDEN + n];
    if (n < 16)
        ws[WS_W2T_OFF + n * HIDDEN + k] =
            (n < N_EDGE_TYPES) ? W2[(size_t)k * N_EDGE_TYPES + n] : 0.0f;
    if (n < N_NODE_TYPES && k < LATENT)
        ws[WS_WNT_OFF + n * LATENT + k] = Wn[(size_t)k * N_NODE_TYPES + n];
}

// ---------------------------------------------------------------------------
// Kernel 1: node_logits = z @ W_node + b_node   [50000 x 32]
// One wave per 16-row tile; K=128 -> 32 wmma steps; 2 N-tiles share the A frag.
// ---------------------------------------------------------------------------
template <bool PRET>
__global__ __launch_bounds__(32) void node_logits_kernel(
    const float* __restrict__ z, const float* __restrict__ Wn,
    const float* __restrict__ WnT, const float* __restrict__ bn,
    float* __restrict__ out) {
    const int lane = threadIdx.x;
    const int m  = lane & 15;   // A row M / B col N / C col N
    const int kh = lane >> 4;   // K-pair selector
    const int rowBase = blockIdx.x * 16;

    v8f c0 = {}; v8f c1 = {};
    const float* zrow = z + (size_t)(rowBase + m) * LATENT;

    for (int kb = 0; kb < LATENT; kb += 4) {
        const int k0 = kb + 2 * kh;
        const v2f a = *(const v2f*)(zrow + k0);
        v2f b0, b1v;
        if constexpr (PRET) {
            b0  = *(const v2f*)(WnT + m * LATENT + k0);
            b1v = *(const v2f*)(WnT + (16 + m) * LATENT + k0);
        } else {
            b0.x  = Wn[(size_t)k0 * N_NODE_TYPES + m];
            b0.y  = Wn[(size_t)(k0 + 1) * N_NODE_TYPES + m];
            b1v.x = Wn[(size_t)k0 * N_NODE_TYPES + 16 + m];
            b1v.y = Wn[(size_t)(k0 + 1) * N_NODE_TYPES + 16 + m];
        }
        c0 = wmma_f32(a, b0, c0);
        c1 = wmma_f32(a, b1v, c1);
    }

    const float bias0 = bn[m];
    const float bias1 = bn[16 + m];
#pragma unroll
    for (int r = 0; r < 8; ++r) {
        const int mr = r + 8 * kh;                  // C row
        float* o = out + (size_t)(rowBase + mr) * N_NODE_TYPES;
        o[m]      = c0[r] + bias0;
        o[16 + m] = c1[r] + bias1;
    }
}

// ---------------------------------------------------------------------------
// Kernel 2: fused edge MLP.
// Block = 256 threads (8 waves) handles 32 edges (two 16-edge M-tiles).
// Gather concat(z[row], z[col]) to LDS (async-to-LDS if available), layer-1
// GEMM (K=256, N=256) via f32 WMMA, bias+ReLU to LDS, layer-2 GEMM (K=256,
// N=8 zero-padded to 16).
// ---------------------------------------------------------------------------
template <bool PRET>
__global__ __launch_bounds__(256) void edge_mlp_kernel(
    const float* __restrict__ z, const int* __restrict__ eidx,
    const float* __restrict__ W1, const float* __restrict__ W1T,
    const float* __restrict__ b1,
    const float* __restrict__ W2, const float* __restrict__ W2T,
    const float* __restrict__ b2,
    float* __restrict__ out) {
    __shared__ float sFeat[32 * FSTRIDE];   // 32 edges x 256 (padded) features
    __shared__ float sH[32 * FSTRIDE];      // 32 edges x 256 (padded) hidden

    const int tid = threadIdx.x;
    const int edgeBase = blockIdx.x * 32;

    // --- Gather features: 32 edges x 64 float4 slots, 8 per thread ---
    for (int i = tid; i < 32 * 64; i += 256) {
        const int e = i >> 6;            // edge within block
        const int q = i & 63;            // float4 slot; 0..31 src half, 32..63 dst half
        const int node = (q < 32) ? eidx[edgeBase + e] : eidx[N_EDGES + edgeBase + e];
        const int off = (q & 31) * 4;    // float offset within 128-float half
        const float* src = z + (size_t)node * LATENT + off;
        float* dst = &sFeat[e * FSTRIDE + ((q < 32) ? 0 : LATENT) + off];
#ifdef USE_ASYNC_LDS
        __builtin_amdgcn_global_load_async_to_lds_b128(
            (g_vi4*)(uintptr_t)src,
            (l_vi4*)(uint32_t)(uintptr_t)dst,
            0, 0);
#else
        const float4 v = *(const float4*)src;
        *(float4*)dst = v;
#endif
    }
#ifdef USE_ASYNC_LDS
    asm volatile("s_wait_asynccnt 0x0" ::: "memory");
#endif
    __syncthreads();

    const int lane  = tid & 31;
    const int wave  = tid >> 5;
    const int m     = lane & 15;
    const int kh    = lane >> 4;
    const int mtile = wave >> 2;    // 0..1 : which 16-edge tile
    const int nslot = wave & 3;     // 0..3 : 4 consecutive 16-wide N-tiles each

    // --- Layer 1: h = relu(feat @ W1 + b1), 4 N-tiles per wave, K=256 ---
    v8f acc[4] = {};
    const float* sA = &sFeat[(mtile * 16 + m) * FSTRIDE];
    for (int kb = 0; kb < 2 * LATENT; kb += 4) {
        const int k0 = kb + 2 * kh;
        const v2f a = *(const v2f*)(sA + k0);
#pragma unroll
        for (int t = 0; t < 4; ++t) {
            const int ncol = (nslot * 4 + t) * 16 + m;
            v2f b;
            if constexpr (PRET) {
                b = *(const v2f*)(W1T + (size_t)ncol * HIDDEN + k0);
            } else {
                b.x = W1[(size_t)k0 * HIDDEN + ncol];
                b.y = W1[(size_t)(k0 + 1) * HIDDEN + ncol];
            }
            acc[t] = wmma_f32(a, b, acc[t]);
        }
    }
#pragma unroll
    for (int t = 0; t < 4; ++t) {
        const int n = (nslot * 4 + t) * 16 + m;     // C col
        const float bias = b1[n];
#pragma unroll
        for (int r = 0; r < 8; ++r) {
            float v = acc[t][r] + bias;
            v = v > 0.0f ? v : 0.0f;
            sH[(mtile * 16 + r + 8 * kh) * FSTRIDE + n] = v;
        }
    }
    __syncthreads();

    // --- Layer 2: logits = h @ W2 + b2 ; one wave per M-tile (wave-uniform) ---
    if (nslot == 0) {
        v8f c2 = {};
        const float* sHA = &sH[(mtile * 16 + m) * FSTRIDE];
        const int nn = (m < N_EDGE_TYPES) ? m : 0;
        const float keep = (m < N_EDGE_TYPES) ? 1.0f : 0.0f;
        for (int kb = 0; kb < HIDDEN; kb += 4) {
            const int k0 = kb + 2 * kh;
            const v2f a = *(const v2f*)(sHA + k0);
            v2f b;
            if constexpr (PRET) {
                b = *(const v2f*)(W2T + (size_t)m * HIDDEN + k0);  // rows 8..15 are 0
            } else {
                b.x = keep * W2[(size_t)k0 * N_EDGE_TYPES + nn];
                b.y = keep * W2[(size_t)(k0 + 1) * N_EDGE_TYPES + nn];
            }
            c2 = wmma_f32(a, b, c2);
        }
        if (m < N_EDGE_TYPES) {   // predicated stores only (no WMMA under divergence)
            const float bias = b2[m];
#pragma unroll
            for (int r = 0; r < 8; ++r) {
                const int mr = r + 8 * kh;
                out[(size_t)(edgeBase + mtile * 16 + mr) * N_EDGE_TYPES + m] =
                    c2[r] + bias;
            }
        }
    }
}

extern "C" void kernel_launch(void* const* d_in, const int* in_sizes, int n_in,
                              void* d_out, int out_size, void* d_ws, size_t ws_size,
                              hipStream_t stream) {
    const float* z    = (const float*)d_in[0];
    const int*   eidx = (const int*)d_in[1];     // [2, N_EDGES] int32
    const float* Wn   = (const float*)d_in[2];
    const float* bn   = (const float*)d_in[3];
    const float* W1   = (const float*)d_in[4];
    const float* b1   = (const float*)d_in[5];
    const float* W2   = (const float*)d_in[6];
    const float* b2   = (const float*)d_in[7];

    float* node_out = (float*)d_out;                               // 50000*32
    float* edge_out = node_out + (size_t)N_NODES * N_NODE_TYPES;   // 800000*8
    float* ws = (float*)d_ws;

    const bool pret = ws_size >= (size_t)WS_FLOATS * sizeof(float);

    if (pret) {
        prep_weights_kernel<<<HIDDEN, HIDDEN, 0, stream>>>(W1, W2, Wn, ws);
        node_logits_kernel<true><<<N_NODES / 16, 32, 0, stream>>>(
            z, Wn, ws + WS_WNT_OFF, bn, node_out);
        edge_mlp_kernel<true><<<N_EDGES / 32, 256, 0, stream>>>(
            z, eidx, W1, ws + WS_W1T_OFF, b1, W2, ws + WS_W2T_OFF, b2, edge_out);
    } else {
        node_logits_kernel<false><<<N_NODES / 16, 32, 0, stream>>>(
            z, Wn, Wn, bn, node_out);
        edge_mlp_kernel<false><<<N_EDGES / 32, 256, 0, stream>>>(
            z, eidx, W1, W1, b1, W2, W2, b2, edge_out);
    }
}